// GraphAutoencoder_3418793967719
// MI455X (gfx1250) — compile-verified
//
#include <hip/hip_runtime.h>
#include <hip/hip_bf16.h>

typedef __attribute__((ext_vector_type(2))) float v2f;
typedef __attribute__((ext_vector_type(8))) float v8f;

namespace {
constexpr int NN   = 50000;
constexpr int NE   = 800000;
constexpr int FIN  = 256;
constexpr int FHID = 128;
constexpr int FEMB = 64;
constexpr int KS   = 2;
constexpr int MT   = 4;   // M-tiles per wave in the WMMA GEMM
}

// ---------------- elementwise / graph helpers ----------------

__global__ void zero4_kernel(float4* __restrict__ p, long long n4) {
  long long i = (long long)blockIdx.x * blockDim.x + threadIdx.x;
  long long stride = (long long)gridDim.x * blockDim.x;
  float4 z = {0.f, 0.f, 0.f, 0.f};
  for (; i < n4; i += stride) p[i] = z;
}

__global__ void degree_kernel(const long long* __restrict__ dst,
                              float* __restrict__ deg, int E) {
  int e = blockIdx.x * blockDim.x + threadIdx.x;
  if (e < E) atomicAdd(&deg[(int)dst[e]], 1.0f);
}

__global__ void dinv_kernel(float* __restrict__ deg, int n) {
  int i = blockIdx.x * blockDim.x + threadIdx.x;
  if (i < n) {
    float d = deg[i];
    deg[i] = (d > 0.0f) ? rsqrtf(d) : 0.0f;
  }
}

__global__ void norm_kernel(const long long* __restrict__ src,
                            const long long* __restrict__ dst,
                            const float* __restrict__ dinv,
                            float* __restrict__ nrm, int E) {
  int e = blockIdx.x * blockDim.x + threadIdx.x;
  if (e < E) nrm[e] = dinv[(int)src[e]] * dinv[(int)dst[e]];
}

// AGG[dst, :] += H[src, :] * norm[e]; row length = 1<<kfShift floats,
// one thread moves 4 floats (float4 gather + 4 fp32 global atomics).
// Edge metadata loads are wave-uniform (all 32 lanes share one edge).
__global__ void scatter_kernel(const float* __restrict__ H,
                               const float* __restrict__ nrm,
                               const long long* __restrict__ src,
                               const long long* __restrict__ dst,
                               float* __restrict__ AGG, int E, int kfShift) {
  long long tid = (long long)blockIdx.x * blockDim.x + threadIdx.x;
  int chunkShift = kfShift - 2;
  long long total = ((long long)E) << chunkShift;
  if (tid >= total) return;
  long long e = tid >> chunkShift;
  int c = (int)(tid & ((1LL << chunkShift) - 1));
  long long s = src[e];
  long long d = dst[e];
  float w = nrm[e];
  const float4 v = ((const float4*)(H + (s << kfShift)))[c];
  float* a = AGG + (d << kfShift) + 4 * c;
  atomicAdd(a + 0, v.x * w);
  atomicAdd(a + 1, v.y * w);
  atomicAdd(a + 2, v.z * w);
  atomicAdd(a + 3, v.w * w);
}

// agg = relu(agg + root + bias[k, f]) over [N, K, F], float4-wide.
__global__ void combine_relu_kernel(float4* __restrict__ agg,
                                    const float4* __restrict__ root,
                                    const float4* __restrict__ bias,
                                    long long total4, int f4Shift) {
  long long i = (long long)blockIdx.x * blockDim.x + threadIdx.x;
  if (i >= total4) return;
  int F4 = 1 << f4Shift;
  int f4 = (int)(i & (F4 - 1));
  int k = (int)((i >> f4Shift) & (KS - 1));
  float4 b = bias[k * F4 + f4];
  float4 v = agg[i];
  float4 r = root[i];
  v.x = fmaxf(v.x + r.x + b.x, 0.0f);
  v.y = fmaxf(v.y + r.y + b.y, 0.0f);
  v.z = fmaxf(v.z + r.z + b.z, 0.0f);
  v.w = fmaxf(v.w + r.w + b.w, 0.0f);
  agg[i] = v;
}

// out[n, f] = relu(0.5 * (in[n,0,f] + in[n,1,f])), float4-wide.
__global__ void mean_relu_kernel(const float4* __restrict__ in,
                                 float4* __restrict__ out,
                                 long long n4, int f4Shift) {
  long long i = (long long)blockIdx.x * blockDim.x + threadIdx.x;
  if (i >= n4) return;
  int F4 = 1 << f4Shift;
  long long n = i >> f4Shift;
  int f4 = (int)(i & (F4 - 1));
  const float4* row = in + (n << (f4Shift + 1));
  float4 a = row[f4];
  float4 b = row[F4 + f4];
  float4 o;
  o.x = fmaxf(0.5f * (a.x + b.x), 0.0f);
  o.y = fmaxf(0.5f * (a.y + b.y), 0.0f);
  o.z = fmaxf(0.5f * (a.z + b.z), 0.0f);
  o.w = fmaxf(0.5f * (a.w + b.w), 0.0f);
  out[i] = o;
}

// ---------------- fp32 WMMA GEMM ----------------
// C[m, n] (+bias, opt relu) = sum_f A[m, f] * W[f, n], per stack k = blockIdx.z.
// Each wave owns ONE 16-wide N-tile and FOUR 16-row M-tiles: the B operand is
// loaded once per k-step and fed to 4 v_wmma_f32_16x16x4_f32 (1.25 loads/WMMA).
// Tail M-group clamps A pointers (wave-uniform; EXEC stays all-ones) and
// guards stores.  VGPR layouts per CDNA5 ISA 7.12.2.
__global__ void gemm_wmma_f32(const float* __restrict__ A,
                              const float* __restrict__ W,
                              float* __restrict__ C,
                              const float* __restrict__ bias,
                              int M, int Fin, int Fout,
                              long long aOffK, int aStride,
                              long long cOffK, int cStride,
                              int wStride, int doRelu) {
  int k = blockIdx.z;
  const float* Ab = A + (long long)k * aOffK;
  const float* Wb = W + (long long)k * wStride;
  float* Cb = C + (long long)k * cOffK;

  int wave = threadIdx.x >> 5;
  int lane = threadIdx.x & 31;
  int nTilesN = Fout >> 4;
  int mtCnt = M >> 4;
  int nGroups = (mtCnt + MT - 1) / MT;
  int wid = blockIdx.x * (blockDim.x >> 5) + wave;
  if (wid >= nGroups * nTilesN) return;   // wave-uniform
  int g  = wid / nTilesN;
  int tn = wid - g * nTilesN;
  int n0 = tn << 4;
  int half = lane >> 4;                   // 0: lanes 0-15, 1: lanes 16-31
  int l16 = lane & 15;

  const float* arow[MT];
  int mt[MT];
#pragma unroll
  for (int i = 0; i < MT; ++i) {
    int t = g * MT + i;
    mt[i] = t;
    int tc = (t < mtCnt) ? t : (mtCnt - 1);
    arow[i] = Ab + (long long)((tc << 4) + l16) * aStride;
  }
  const float* wcol = Wb + n0 + l16;

  v8f acc0 = {0.f, 0.f, 0.f, 0.f, 0.f, 0.f, 0.f, 0.f};
  v8f acc1 = acc0, acc2 = acc0, acc3 = acc0;

  for (int kk = 0; kk < Fin; kk += 4) {
    int kb = kk + 2 * half;
    v2f b;
    b.x = wcol[(long long)(kb + 0) * Fout];
    b.y = wcol[(long long)(kb + 1) * Fout];
    v2f a0, a1, a2, a3;
    a0.x = arow[0][kb]; a0.y = arow[0][kb + 1];
    a1.x = arow[1][kb]; a1.y = arow[1][kb + 1];
    a2.x = arow[2][kb]; a2.y = arow[2][kb + 1];
    a3.x = arow[3][kb]; a3.y = arow[3][kb + 1];
    acc0 = __builtin_amdgcn_wmma_f32_16x16x4_f32(false, a0, false, b,
                                                 (short)0, acc0, false, false);
    acc1 = __builtin_amdgcn_wmma_f32_16x16x4_f32(false, a1, false, b,
                                                 (short)0, acc1, false, false);
    acc2 = __builtin_amdgcn_wmma_f32_16x16x4_f32(false, a2, false, b,
                                                 (short)0, acc2, false, false);
    acc3 = __builtin_amdgcn_wmma_f32_16x16x4_f32(false, a3, false, b,
                                                 (short)0, acc3, false, false);
  }

  float bv = bias ? bias[(long long)k * Fout + n0 + l16] : 0.0f;
  float* ccol = Cb + n0 + l16;
  auto store_tile = [&](const v8f& acc, int t) {
    if (t < mtCnt) {
      int m0 = t << 4;
#pragma unroll
      for (int gg = 0; gg < 8; ++gg) {
        float v = acc[gg] + bv;
        if (doRelu) v = (v > 0.0f) ? v : 0.0f;
        ccol[(long long)(m0 + gg + 8 * half) * cStride] = v;
      }
    }
  };
  store_tile(acc0, mt[0]);
  store_tile(acc1, mt[1]);
  store_tile(acc2, mt[2]);
  store_tile(acc3, mt[3]);
}

// ---------------- host orchestration ----------------

extern "C" void kernel_launch(void* const* d_in, const int* in_sizes, int n_in,
                              void* d_out, int out_size, void* d_ws, size_t ws_size,
                              hipStream_t stream) {
  const float*     x       = (const float*)d_in[0];
  const long long* ei      = (const long long*)d_in[1];   // int64 [2, E]
  const float*     init_w1 = (const float*)d_in[2];
  const float*     w1      = (const float*)d_in[3];
  const float*     root_w1 = (const float*)d_in[4];
  const float*     b1      = (const float*)d_in[5];
  const float*     init_w2 = (const float*)d_in[6];
  const float*     w2      = (const float*)d_in[7];
  const float*     root_w2 = (const float*)d_in[8];
  const float*     b2      = (const float*)d_in[9];
  const float*     fc1_w   = (const float*)d_in[10];
  const float*     fc1_b   = (const float*)d_in[11];
  const float*     fc2_w   = (const float*)d_in[12];
  const float*     fc2_b   = (const float*)d_in[13];

  const long long* src = ei;
  const long long* dst = ei + NE;

  // workspace layout (floats), regions reused across phases
  float* ws  = (float*)d_ws;
  float* deg = ws;                                        // NN (deg -> dinv)
  float* nrm = ws + NN;                                   // NE
  long long nkh = (long long)NN * KS * FHID;              // 12.8M
  long long nke = (long long)NN * KS * FEMB;              // 6.4M
  float* R1 = nrm + NE;                                   // nkh
  float* X1 = R1 + nkh;                                   // nkh
  float* X2 = X1 + nkh;                                   // nkh
  // reuse after conv1:
  float* h1   = R1;                                       // NN*FHID
  float* dbuf = R1 + (long long)NN * FHID;                // NN*FHID
  float* R2   = X1;                                       // nke
  float* Y1   = X1 + nke;                                 // nke
  float* Y2   = X2;                                       // nke
  float* lat  = X2 + nke;                                 // NN*FEMB

  auto gemm = [&](const float* A, const float* W, float* C, const float* bias,
                  int M, int Fin, int Fout, long long aOffK, int aStride,
                  long long cOffK, int cStride, int wStrideK, int relu, int kz) {
    int mtCnt  = M / 16;
    int groups = (mtCnt + MT - 1) / MT;
    int waves  = groups * (Fout / 16);
    int blocks = (waves + 3) / 4;                          // 4 waves / block
    gemm_wmma_f32<<<dim3(blocks, 1, kz), 128, 0, stream>>>(
        A, W, C, bias, M, Fin, Fout, aOffK, aStride, cOffK, cStride,
        wStrideK, relu);
  };

  // -------- gcn_norm --------
  zero4_kernel<<<64, 256, 0, stream>>>((float4*)deg, (long long)NN / 4);
  degree_kernel<<<(NE + 255) / 256, 256, 0, stream>>>(dst, deg, NE);
  dinv_kernel<<<(NN + 255) / 256, 256, 0, stream>>>(deg, NN);
  norm_kernel<<<(NE + 255) / 256, 256, 0, stream>>>(src, dst, deg, nrm, NE);

  // -------- ARMA conv 1: FIN -> FHID --------
  gemm(x, root_w1, R1, nullptr, NN, FIN, FHID, 0, FIN, FHID, KS * FHID,
       FIN * FHID, 0, KS);
  gemm(x, init_w1, X1, nullptr, NN, FIN, FHID, 0, FIN, FHID, KS * FHID,
       FIN * FHID, 0, KS);
  for (int t = 0; t < 3; ++t) {
    if (t > 0)
      gemm(X2, w1, X1, nullptr, NN, FHID, FHID, FHID, KS * FHID, FHID,
           KS * FHID, FHID * FHID, 0, KS);
    zero4_kernel<<<4096, 256, 0, stream>>>((float4*)X2, nkh / 4);
    long long st = (long long)NE << 6;                    // E * (256/4)
    scatter_kernel<<<(unsigned)((st + 255) / 256), 256, 0, stream>>>(
        X1, nrm, src, dst, X2, NE, 8);
    combine_relu_kernel<<<(unsigned)((nkh / 4 + 255) / 256), 256, 0, stream>>>(
        (float4*)X2, (const float4*)R1, (const float4*)b1, nkh / 4, 5);
  }
  mean_relu_kernel<<<(unsigned)(((long long)NN * FHID / 4 + 255) / 256), 256, 0,
                    stream>>>((const float4*)X2, (float4*)h1,
                              (long long)NN * FHID / 4, 5);

  // -------- ARMA conv 2: FHID -> FEMB --------
  gemm(h1, root_w2, R2, nullptr, NN, FHID, FEMB, 0, FHID, FEMB, KS * FEMB,
       FHID * FEMB, 0, KS);
  gemm(h1, init_w2, Y1, nullptr, NN, FHID, FEMB, 0, FHID, FEMB, KS * FEMB,
       FHID * FEMB, 0, KS);
  for (int t = 0; t < 3; ++t) {
    if (t > 0)
      gemm(Y2, w2, Y1, nullptr, NN, FEMB, FEMB, FEMB, KS * FEMB, FEMB,
           KS * FEMB, FEMB * FEMB, 0, KS);
    zero4_kernel<<<4096, 256, 0, stream>>>((float4*)Y2, nke / 4);
    long long st = (long long)NE << 5;                    // E * (128/4)
    scatter_kernel<<<(unsigned)((st + 255) / 256), 256, 0, stream>>>(
        Y1, nrm, src, dst, Y2, NE, 7);
    combine_relu_kernel<<<(unsigned)((nke / 4 + 255) / 256), 256, 0, stream>>>(
        (float4*)Y2, (const float4*)R2, (const float4*)b2, nke / 4, 4);
  }
  mean_relu_kernel<<<(unsigned)(((long long)NN * FEMB / 4 + 255) / 256), 256, 0,
                    stream>>>((const float4*)Y2, (float4*)lat,
                              (long long)NN * FEMB / 4, 4);

  // -------- decoder MLP --------
  gemm(lat, fc1_w, dbuf, fc1_b, NN, FEMB, FHID, 0, FEMB, 0, FHID, 0, 1, 1);
  gemm(dbuf, fc2_w, (float*)d_out, fc2_b, NN, FHID, FIN, 0, FHID, 0, FIN, 0, 0,
       1);
}